// MultiBoxLoss_30142080483607
// MI455X (gfx1250) — compile-verified
//
#include <hip/hip_runtime.h>
#include <hip/hip_bf16.h>
#include <stdint.h>

#define B_ 64
#define P_ 8732
#define M_ 32
#define C_ 21
#define NCHUNK 35            // ceil(8732/256)
#define THRESH_ 0.5f
#define VAR0_ 0.1f
#define VAR1_ 0.2f

typedef float v2f __attribute__((ext_vector_type(2)));
typedef float v8f __attribute__((ext_vector_type(8)));

__device__ __forceinline__ float sl1(float x) {
    float a = fabsf(x);
    return a < 1.0f ? 0.5f * a * a : a - 0.5f;
}

// Full-wave (32-lane) sum. Uses V_WMMA_F32_16X16X4_F32 with B = ones:
// A[m][0] = v_m (lanes 0-15), A[m][2] = v_{m+16} (lanes 16-31), other K slots 0.
// D[m][n] = v_m + v_{m+16} (replicated over n). Each lane sums its 8 D regs
// (= rows 0-7 total for lanes<16, rows 8-15 total for lanes>=16); one xor-16
// shuffle completes the 32-lane sum in every lane. EXEC must be all-ones at
// the call site (callers keep all 256 threads alive).
__device__ __forceinline__ float wave_reduce_sum(float v) {
#if __has_builtin(__builtin_amdgcn_wmma_f32_16x16x4_f32)
    v2f a;  a[0] = v;    a[1] = 0.0f;
    v2f bm; bm[0] = 1.0f; bm[1] = 1.0f;   // all-ones B: striping-agnostic
    v8f c = {0.f, 0.f, 0.f, 0.f, 0.f, 0.f, 0.f, 0.f};
    v8f d = __builtin_amdgcn_wmma_f32_16x16x4_f32(false, a, false, bm,
                                                  (short)0, c, false, false);
    float s = d[0] + d[1] + d[2] + d[3] + d[4] + d[5] + d[6] + d[7];
    s += __shfl_xor(s, 16, 32);
    return s;
#else
    for (int off = 16; off; off >>= 1) v += __shfl_xor(v, off, 32);
    return v;
#endif
}

// ---------------------------------------------------------------------------
// Kernel 1: per-batch matching. One block per batch, 256 threads loop priors.
// Outputs per-prior best_truth_ov / best_truth_idx, and the 32 force-match
// prior indices (per-truth argmax over priors, first-index tie-break).
// ---------------------------------------------------------------------------
__global__ __launch_bounds__(256) void k_match(const float* __restrict__ priors,
                                               const float* __restrict__ targets,
                                               float* __restrict__ best_ov,
                                               int*   __restrict__ best_idx,
                                               int*   __restrict__ force_p) {
    const int b   = blockIdx.x;
    const int tid = threadIdx.x;
    __shared__ float tb[M_][4];
    __shared__ float tarea[M_];
    __shared__ unsigned long long bestkey[M_];

    if (tid < M_) {
        const float* t = targets + ((size_t)b * M_ + tid) * 5;
        float x1 = t[0], y1 = t[1], x2 = t[2], y2 = t[3];
        tb[tid][0] = x1; tb[tid][1] = y1; tb[tid][2] = x2; tb[tid][3] = y2;
        tarea[tid] = (x2 - x1) * (y2 - y1);
        bestkey[tid] = 0ull;
    }
    __syncthreads();

    unsigned long long mykey[M_];
#pragma unroll
    for (int t = 0; t < M_; ++t) mykey[t] = 0ull;

    for (int p = tid; p < P_; p += 256) {
        float4 pr = ((const float4*)priors)[p];
        float px1 = pr.x - pr.z * 0.5f, py1 = pr.y - pr.w * 0.5f;
        float px2 = pr.x + pr.z * 0.5f, py2 = pr.y + pr.w * 0.5f;
        float parea = pr.z * pr.w;
        float bov = -1.0f; int bidx = 0;
#pragma unroll
        for (int t = 0; t < M_; ++t) {
            float lx = fmaxf(tb[t][0], px1), ly = fmaxf(tb[t][1], py1);
            float rx = fminf(tb[t][2], px2), ry = fminf(tb[t][3], py2);
            float w = fmaxf(rx - lx, 0.0f), h = fmaxf(ry - ly, 0.0f);
            float inter = w * h;
            float iou = inter / (tarea[t] + parea - inter);
            if (iou > bov) { bov = iou; bidx = t; }   // first-index tie-break
            unsigned long long key =
                ((unsigned long long)__float_as_uint(iou) << 32) |
                (unsigned)(~(unsigned)p);             // smaller p wins ties
            if (key > mykey[t]) mykey[t] = key;
        }
        best_ov [(size_t)b * P_ + p] = bov;
        best_idx[(size_t)b * P_ + p] = bidx;
    }
#pragma unroll
    for (int t = 0; t < M_; ++t) atomicMax(&bestkey[t], mykey[t]);   // ds_max_u64
    __syncthreads();
    if (tid < M_) {
        unsigned p = ~(unsigned)(bestkey[tid] & 0xFFFFFFFFull);
        force_p[b * M_ + tid] = (int)p;
    }
}

// ---------------------------------------------------------------------------
// Kernel 2: encode + smooth-L1 + CE (logsumexp). conf rows are staged into
// LDS with async global->LDS b128 copies (ASYNCcnt path). Per-block sums are
// reduced with the WMMA wave reduction and written to fixed slots.
// ---------------------------------------------------------------------------
__global__ __launch_bounds__(256) void k_loss(const float* __restrict__ loc,
                                              const float* __restrict__ conf,
                                              const float* __restrict__ priors,
                                              const float* __restrict__ targets,
                                              const float* __restrict__ best_ov,
                                              const int*   __restrict__ best_idx,
                                              const int*   __restrict__ force_p,
                                              float* __restrict__ negval,
                                              int*   __restrict__ num_pos,
                                              float* __restrict__ blk_ll,
                                              float* __restrict__ blk_ce) {
    const int b     = blockIdx.x / NCHUNK;
    const int chunk = blockIdx.x % NCHUNK;
    const int p0    = chunk << 8;
    const int tid   = threadIdx.x;
    const int p     = p0 + tid;

    __shared__ __align__(16) float tile[256 * C_];   // 21504 B, 16B-aligned
    __shared__ int force_s[M_];
    __shared__ float red_ll[8], red_ce[8];
    __shared__ int red_np[8];

    if (tid < M_) force_s[tid] = force_p[b * M_ + tid];

    // Async-stage conf[b, p0 : p0+rows, :] into LDS. rows*84 bytes is always a
    // multiple of 16 here (rows = 256 or 28).
    {
        int rows  = P_ - p0; if (rows > 256) rows = 256;
        int nvec  = (rows * (C_ * 4)) >> 4;          // 16B units
        unsigned long long gaddr =
            (unsigned long long)(uintptr_t)(conf + ((size_t)b * P_ + p0) * C_);
        unsigned lds_base = (unsigned)(uintptr_t)(void*)tile;  // low 32b = LDS offset
        for (int i = tid; i < nvec; i += 256) {
            unsigned voff = (unsigned)(i << 4);
            asm volatile("global_load_async_to_lds_b128 %0, %1, %2"
                         :
                         : "v"(lds_base + voff), "v"(voff), "s"(gaddr)
                         : "memory");
        }
        asm volatile("s_wait_asynccnt 0" ::: "memory");
    }
    __syncthreads();

    float ll = 0.0f, ce_pos = 0.0f;
    int ispos = 0;
    if (p < P_) {
        size_t bp = (size_t)b * P_ + p;
        float bov = best_ov[bp];
        int   ti  = best_idx[bp];
        // force-match override, last j wins (matches sequential .at[].set)
#pragma unroll
        for (int j = 0; j < M_; ++j)
            if (force_s[j] == p) { bov = 2.0f; ti = j; }

        const float* tg = targets + ((size_t)b * M_ + ti) * 5;
        float mx1 = tg[0], my1 = tg[1], mx2 = tg[2], my2 = tg[3];
        int conf_t = (bov < THRESH_) ? 0 : ((int)tg[4] + 1);
        bool pos = conf_t > 0;

        float4 pr = ((const float4*)priors)[p];
        float gcx = ((mx1 + mx2) * 0.5f - pr.x) / (VAR0_ * pr.z);
        float gcy = ((my1 + my2) * 0.5f - pr.y) / (VAR0_ * pr.w);
        float gw  = logf((mx2 - mx1) / pr.z) / VAR1_;
        float gh  = logf((my2 - my1) / pr.w) / VAR1_;
        float4 ld = ((const float4*)loc)[bp];
        if (pos)
            ll = sl1(ld.x - gcx) + sl1(ld.y - gcy) + sl1(ld.z - gw) + sl1(ld.w - gh);

        const float* row = tile + tid * C_;
        float mx = row[0];
#pragma unroll
        for (int k = 1; k < C_; ++k) mx = fmaxf(mx, row[k]);
        float s = 0.0f;
#pragma unroll
        for (int k = 0; k < C_; ++k) s += expf(row[k] - mx);
        float ce = (mx + logf(s)) - row[conf_t];

        negval[bp] = pos ? 0.0f : ce;
        if (pos) { ce_pos = ce; ispos = 1; }
    }

    // full wave active here -> WMMA reduction legal (EXEC all-ones)
    float wll = wave_reduce_sum(ll);
    float wce = wave_reduce_sum(ce_pos);
    int wnp = (int)__popcll(__ballot(ispos));

    int wave = tid >> 5, lane = tid & 31;
    if (lane == 0) { red_ll[wave] = wll; red_ce[wave] = wce; red_np[wave] = wnp; }
    __syncthreads();
    if (tid == 0) {
        float a = 0.0f, c = 0.0f; int n = 0;
        for (int w = 0; w < 8; ++w) { a += red_ll[w]; c += red_ce[w]; n += red_np[w]; }
        blk_ll[blockIdx.x] = a;
        blk_ce[blockIdx.x] = c;
        if (n) atomicAdd(&num_pos[b], n);
    }
}

// ---------------------------------------------------------------------------
// Kernel 3: per-batch hard-negative mining via 4-pass radix select (MSB-first
// byte histograms on the monotone uint key of the non-negative CE values),
// then a sum pass. neg contribution = sum(key>T) + (K - count(key>T)) * T.
// ---------------------------------------------------------------------------
__global__ __launch_bounds__(256) void k_select(const float* __restrict__ negval,
                                                const int*   __restrict__ num_pos,
                                                float* __restrict__ neg_b) {
    const int b   = blockIdx.x;
    const int tid = threadIdx.x;
    __shared__ unsigned hist[256];
    __shared__ unsigned sh_prefix;
    __shared__ int sh_r;
    __shared__ float s_s[8];
    __shared__ int   s_c[8];

    long long Kl = 3ll * (long long)num_pos[b];
    int K = (Kl > (long long)(P_ - 1)) ? (P_ - 1) : (int)Kl;
    if (K <= 0) { if (tid == 0) neg_b[b] = 0.0f; return; }

    const float* vals = negval + (size_t)b * P_;
    unsigned prefix = 0;
    int r = K;

    for (int pass = 0; pass < 4; ++pass) {
        const int shift = 24 - pass * 8;
        hist[tid] = 0u;
        __syncthreads();
        for (int p = tid; p < P_; p += 256) {
            unsigned key = __float_as_uint(vals[p]);
            bool match = (pass == 0) || ((key >> (shift + 8)) == prefix);
            if (match) atomicAdd(&hist[(key >> shift) & 0xFFu], 1u);
        }
        __syncthreads();
        if (tid == 0) {
            int acc = 0; unsigned bin = 0;
            for (int i = 255; i >= 0; --i) {
                acc += (int)hist[i];
                if (acc >= r) { bin = (unsigned)i; r -= (acc - (int)hist[i]); break; }
            }
            sh_prefix = (prefix << 8) | bin;
            sh_r = r;
        }
        __syncthreads();
        prefix = sh_prefix;
        r = sh_r;
    }

    const unsigned Tk = prefix;            // key of the K-th largest value
    float ssum = 0.0f; int cgt = 0;
    for (int p = tid; p < P_; p += 256) {
        float v = vals[p];
        if (__float_as_uint(v) > Tk) { ssum += v; ++cgt; }
    }
    for (int off = 16; off; off >>= 1) {
        ssum += __shfl_down(ssum, off, 32);
        cgt  += __shfl_down(cgt,  off, 32);
    }
    if ((tid & 31) == 0) { s_s[tid >> 5] = ssum; s_c[tid >> 5] = cgt; }
    __syncthreads();
    if (tid == 0) {
        float S = 0.0f; int Ct = 0;
        for (int w = 0; w < 8; ++w) { S += s_s[w]; Ct += s_c[w]; }
        neg_b[b] = S + (float)(K - Ct) * __uint_as_float(Tk);
    }
}

// ---------------------------------------------------------------------------
// Kernel 4: deterministic final reduction and division by N.
// ---------------------------------------------------------------------------
__global__ __launch_bounds__(256) void k_final(const float* __restrict__ blk_ll,
                                               const float* __restrict__ blk_ce,
                                               const float* __restrict__ neg_b,
                                               const int*   __restrict__ num_pos,
                                               int nblk,
                                               float* __restrict__ out) {
    const int tid = threadIdx.x;
    __shared__ float r1[256], r2[256], r3[256];
    __shared__ int   r4[256];
    float ll = 0.0f, ce = 0.0f, ng = 0.0f; int np = 0;
    for (int i = tid; i < nblk; i += 256) { ll += blk_ll[i]; ce += blk_ce[i]; }
    if (tid < B_) { ng = neg_b[tid]; np = num_pos[tid]; }
    r1[tid] = ll; r2[tid] = ce; r3[tid] = ng; r4[tid] = np;
    __syncthreads();
    for (int s = 128; s > 0; s >>= 1) {
        if (tid < s) {
            r1[tid] += r1[tid + s]; r2[tid] += r2[tid + s];
            r3[tid] += r3[tid + s]; r4[tid] += r4[tid + s];
        }
        __syncthreads();
    }
    if (tid == 0) {
        float N = (float)r4[0];
        out[0] = r1[0] / N;
        out[1] = (r2[0] + r3[0]) / N;
    }
}

// ---------------------------------------------------------------------------
extern "C" void kernel_launch(void* const* d_in, const int* in_sizes, int n_in,
                              void* d_out, int out_size, void* d_ws, size_t ws_size,
                              hipStream_t stream) {
    (void)in_sizes; (void)n_in; (void)out_size; (void)ws_size;
    const float* loc     = (const float*)d_in[0];   // (64, 8732, 4)
    const float* conf    = (const float*)d_in[1];   // (64, 8732, 21)
    const float* priors  = (const float*)d_in[2];   // (8732, 4)
    const float* targets = (const float*)d_in[3];   // (64, 32, 5)

    char* ws = (char*)d_ws;
    const size_t SP = (size_t)B_ * P_ * sizeof(float);   // 2,235,392 B
    float* best_ov  = (float*)(ws);
    int*   best_idx = (int*)  (ws + SP);
    float* negval   = (float*)(ws + 2 * SP);
    int*   num_pos  = (int*)  (ws + 3 * SP);                         // 256 B slot
    float* blk_ll   = (float*)(ws + 3 * SP + 256);                   // 2240 f
    float* blk_ce   = (float*)(ws + 3 * SP + 256 + 9216);            // 2240 f
    float* neg_b    = (float*)(ws + 3 * SP + 256 + 2 * 9216);        // 64 f
    int*   force_p  = (int*)  (ws + 3 * SP + 256 + 2 * 9216 + 256);  // 64*32 i

    hipMemsetAsync(num_pos, 0, B_ * sizeof(int), stream);

    k_match <<<B_,          256, 0, stream>>>(priors, targets, best_ov, best_idx, force_p);
    k_loss  <<<B_ * NCHUNK, 256, 0, stream>>>(loc, conf, priors, targets, best_ov,
                                              best_idx, force_p, negval, num_pos,
                                              blk_ll, blk_ce);
    k_select<<<B_,          256, 0, stream>>>(negval, num_pos, neg_b);
    k_final <<<1,           256, 0, stream>>>(blk_ll, blk_ce, neg_b, num_pos,
                                              B_ * NCHUNK, (float*)d_out);
}